// GCN_39977555591218
// MI455X (gfx1250) — compile-verified
//
#include <hip/hip_runtime.h>
#include <hip/hip_bf16.h>

typedef __attribute__((ext_vector_type(2))) float v2f;
typedef __attribute__((ext_vector_type(8))) float v8f;

#define N_NODES 50000
#define N_EDGES 800000
#define IN_F 1024
#define H_F 96
#define C_F 40

// ---------------------------------------------------------------------------
// GEMM1: XW[N,96] = X[N,1024] @ W1[1024,96]   (f32 WMMA 16x16x4)
// Block = 128 threads (4 waves). Block tile: 128 rows x 96 cols.
// Each wave: 32 rows (2 M-tiles) x 6 N-tiles = 12 accumulators.
// B staged k-pair interleaved so each B frag is a single aligned ds_load_b64.
// ---------------------------------------------------------------------------
#define AS1_STRIDE 36    // 32 k + pad; 36*r mod 64 distinct for r=0..15
#define BS1_STRIDE 224   // 96 cols * 2 (k-pair) + 32 pad; 224 mod 64 == 32

__global__ __launch_bounds__(128) void gcn_gemm1_wmma(const float* __restrict__ X,
                                                      const float* __restrict__ W1,
                                                      float* __restrict__ XW) {
  __shared__ float As[128 * AS1_STRIDE];   // 18432 B
  __shared__ float Bs[16 * BS1_STRIDE];    // 14336 B
  const int tid = threadIdx.x;
  const int wave = tid >> 5;
  const int lane = tid & 31;
  const int lane16 = lane & 15;
  const int half = lane >> 4;
  const int rowBase = blockIdx.x * 128;

  v8f acc[2][6] = {};

  for (int kBase = 0; kBase < IN_F; kBase += 32) {
    // Stage A: 128 rows x 32 k  (1024 float4, 8 per thread, coalesced along k)
#pragma unroll
    for (int it = 0; it < 8; ++it) {
      int i = tid + it * 128;
      int row = i >> 3;
      int k4 = i & 7;
      int gRow = rowBase + row;
      float4 v = make_float4(0.f, 0.f, 0.f, 0.f);
      if (gRow < N_NODES)
        v = *(const float4*)(X + (size_t)gRow * IN_F + kBase + k4 * 4);
      float* d = &As[row * AS1_STRIDE + k4 * 4];
      d[0] = v.x; d[1] = v.y; d[2] = v.z; d[3] = v.w;
    }
    // Stage B: 32 k x 96 n, k-pair interleaved: Bs[(k/2)*S + n*2 + (k&1)]
#pragma unroll
    for (int it = 0; it < 6; ++it) {
      int i = tid + it * 128;
      int k = i / 24;
      int n4 = i % 24;
      float4 v = *(const float4*)(W1 + (size_t)(kBase + k) * H_F + n4 * 4);
      float* d = &Bs[(k >> 1) * BS1_STRIDE + n4 * 8 + (k & 1)];
      d[0] = v.x; d[2] = v.y; d[4] = v.z; d[6] = v.w;
    }
    __syncthreads();

#pragma unroll
    for (int kk = 0; kk < 8; ++kk) {
      const int k0 = kk * 4 + half * 2;     // lane-half K split per ISA layout
      const int kprow = kk * 2 + half;      // k-pair row in Bs
      v2f a0 = *(const v2f*)&As[(wave * 32 + lane16) * AS1_STRIDE + k0];
      v2f a1 = *(const v2f*)&As[(wave * 32 + 16 + lane16) * AS1_STRIDE + k0];
#pragma unroll
      for (int nt = 0; nt < 6; ++nt) {
        v2f b = *(const v2f*)&Bs[kprow * BS1_STRIDE + (nt * 16 + lane16) * 2];
        acc[0][nt] = __builtin_amdgcn_wmma_f32_16x16x4_f32(
            false, a0, false, b, (short)0, acc[0][nt], false, false);
        acc[1][nt] = __builtin_amdgcn_wmma_f32_16x16x4_f32(
            false, a1, false, b, (short)0, acc[1][nt], false, false);
      }
    }
    __syncthreads();
  }

  // D layout: lane holds col=lane&15; vgpr v -> row v + half*8
#pragma unroll
  for (int mt = 0; mt < 2; ++mt) {
#pragma unroll
    for (int nt = 0; nt < 6; ++nt) {
#pragma unroll
      for (int v = 0; v < 8; ++v) {
        int row = rowBase + wave * 32 + mt * 16 + v + half * 8;
        int col = nt * 16 + lane16;
        if (row < N_NODES) XW[(size_t)row * H_F + col] = acc[mt][nt][v];
      }
    }
  }
}

// ---------------------------------------------------------------------------
// GEMM2: HW[N,40] = H[N,96] @ W2[96,40]  (pad N to 48; 3 n-tiles)
// Block = 128 threads (4 waves), block tile 64 rows x 48 cols.
// ---------------------------------------------------------------------------
#define AS2_STRIDE 100   // 96 + pad; 100*r mod 64 distinct for r=0..15
#define BS2_STRIDE 96    // 48 cols * 2 (k-pair); 96 mod 64 == 32

__global__ __launch_bounds__(128) void gcn_gemm2_wmma(const float* __restrict__ H,
                                                      const float* __restrict__ W2,
                                                      float* __restrict__ HW) {
  __shared__ float As[64 * AS2_STRIDE];    // 25600 B
  __shared__ float Bs[48 * BS2_STRIDE];    // 18432 B
  const int tid = threadIdx.x;
  const int wave = tid >> 5;
  const int lane = tid & 31;
  const int lane16 = lane & 15;
  const int half = lane >> 4;
  const int rowBase = blockIdx.x * 64;

  // Stage W2 k-pair interleaved (96x40, pad cols 40..47 with zeros)
  for (int i = tid; i < 96 * 40; i += 128) {
    int k = i / 40, n = i % 40;
    Bs[(k >> 1) * BS2_STRIDE + n * 2 + (k & 1)] = W2[i];
  }
  for (int i = tid; i < 96 * 8; i += 128) {
    int k = i / 8, n = 40 + (i % 8);
    Bs[(k >> 1) * BS2_STRIDE + n * 2 + (k & 1)] = 0.0f;
  }

  // Stage H tile: 64 rows x 96 k (1536 float4, 12 per thread)
#pragma unroll
  for (int it = 0; it < 12; ++it) {
    int i = tid + it * 128;
    int row = i / 24;
    int n4 = i % 24;
    int gRow = rowBase + row;
    float4 v = make_float4(0.f, 0.f, 0.f, 0.f);
    if (gRow < N_NODES)
      v = *(const float4*)(H + (size_t)gRow * H_F + n4 * 4);
    float* d = &As[row * AS2_STRIDE + n4 * 4];
    d[0] = v.x; d[1] = v.y; d[2] = v.z; d[3] = v.w;
  }
  __syncthreads();

  v8f acc[3] = {};
#pragma unroll
  for (int kk = 0; kk < 24; ++kk) {
    const int k0 = kk * 4 + half * 2;
    const int kprow = kk * 2 + half;
    v2f a = *(const v2f*)&As[(wave * 16 + lane16) * AS2_STRIDE + k0];
#pragma unroll
    for (int nt = 0; nt < 3; ++nt) {
      v2f b = *(const v2f*)&Bs[kprow * BS2_STRIDE + (nt * 16 + lane16) * 2];
      acc[nt] = __builtin_amdgcn_wmma_f32_16x16x4_f32(
          false, a, false, b, (short)0, acc[nt], false, false);
    }
  }

#pragma unroll
  for (int nt = 0; nt < 3; ++nt) {
#pragma unroll
    for (int v = 0; v < 8; ++v) {
      int row = rowBase + wave * 16 + v + half * 8;
      int col = nt * 16 + lane16;
      if (row < N_NODES && col < C_F) HW[(size_t)row * C_F + col] = acc[nt][v];
    }
  }
}

// ---------------------------------------------------------------------------
// Degree / normalization / propagation kernels
// ---------------------------------------------------------------------------
__global__ void gcn_deg_init(float* deg) {
  int i = blockIdx.x * blockDim.x + threadIdx.x;
  if (i < N_NODES) deg[i] = 1.0f;  // self-loop contributes 1
}

__global__ void gcn_deg_accum(const long long* __restrict__ ei, float* deg) {
  int e = blockIdx.x * blockDim.x + threadIdx.x;
  if (e < N_EDGES) atomicAdd(&deg[(int)ei[N_EDGES + e]], 1.0f);
}

__global__ void gcn_deg_to_dinv(float* deg) {
  int i = blockIdx.x * blockDim.x + threadIdx.x;
  if (i < N_NODES) deg[i] = rsqrtf(deg[i]);  // deg >= 1 always
}

__global__ void gcn_zero(float* __restrict__ p, long long n) {
  long long i = (long long)blockIdx.x * blockDim.x + threadIdx.x;
  if (i < n) p[i] = 0.0f;
}

// One thread per (edge, feature): acc[dst,f] += dinv[src]*dinv[dst]*feat[src,f]
__global__ void gcn_scatter(const long long* __restrict__ ei,
                            const float* __restrict__ dinv,
                            const float* __restrict__ feat,
                            float* __restrict__ acc, int F, long long total) {
  long long idx = (long long)blockIdx.x * blockDim.x + threadIdx.x;
  if (idx >= total) return;
  int e = (int)(idx / F);
  int f = (int)(idx % F);
  int s = (int)ei[e];
  int d = (int)ei[N_EDGES + e];
  float nrm = dinv[s] * dinv[d];
  atomicAdd(&acc[(size_t)d * F + f], nrm * feat[(size_t)s * F + f]);
}

// acc[i,f] = (acc[i,f] + dinv[i]^2 * feat[i,f] + bias[f])  [optional relu]
__global__ void gcn_combine(const float* __restrict__ dinv,
                            const float* __restrict__ feat,
                            const float* __restrict__ bias,
                            float* __restrict__ acc, int F, int do_relu,
                            long long total) {
  long long idx = (long long)blockIdx.x * blockDim.x + threadIdx.x;
  if (idx >= total) return;
  int i = (int)(idx / F);
  int f = (int)(idx % F);
  float di = dinv[i];
  float v = acc[idx] + di * di * feat[idx] + bias[f];
  acc[idx] = do_relu ? fmaxf(v, 0.0f) : v;
}

// ---------------------------------------------------------------------------
extern "C" void kernel_launch(void* const* d_in, const int* in_sizes, int n_in,
                              void* d_out, int out_size, void* d_ws, size_t ws_size,
                              hipStream_t stream) {
  const float* X      = (const float*)d_in[0];
  const long long* EI = (const long long*)d_in[1];
  const float* W1     = (const float*)d_in[2];
  const float* b1     = (const float*)d_in[3];
  const float* W2     = (const float*)d_in[4];
  const float* b2     = (const float*)d_in[5];
  float* out = (float*)d_out;
  float* ws  = (float*)d_ws;

  float* dinv = ws;                        // 50048
  float* xw   = dinv + 50048;              // 4,800,000
  float* h    = xw + (size_t)N_NODES * H_F;// 4,800,000 (acc1, then H in-place)
  float* hw   = h + (size_t)N_NODES * H_F; // 2,000,000

  const int T = 256;

  // degrees -> dinv
  gcn_deg_init<<<(N_NODES + T - 1) / T, T, 0, stream>>>(dinv);
  gcn_deg_accum<<<(N_EDGES + T - 1) / T, T, 0, stream>>>(EI, dinv);
  gcn_deg_to_dinv<<<(N_NODES + T - 1) / T, T, 0, stream>>>(dinv);

  // layer 1
  gcn_gemm1_wmma<<<(N_NODES + 127) / 128, 128, 0, stream>>>(X, W1, xw);
  long long n1 = (long long)N_NODES * H_F;
  gcn_zero<<<(unsigned)((n1 + T - 1) / T), T, 0, stream>>>(h, n1);
  long long sc1 = (long long)N_EDGES * H_F;
  gcn_scatter<<<(unsigned)((sc1 + T - 1) / T), T, 0, stream>>>(EI, dinv, xw, h, H_F, sc1);
  gcn_combine<<<(unsigned)((n1 + T - 1) / T), T, 0, stream>>>(dinv, xw, b1, h, H_F, 1, n1);

  // layer 2
  gcn_gemm2_wmma<<<(N_NODES + 63) / 64, 128, 0, stream>>>(h, W2, hw);
  long long n2 = (long long)N_NODES * C_F;
  gcn_zero<<<(unsigned)((n2 + T - 1) / T), T, 0, stream>>>(out, n2);
  long long sc2 = (long long)N_EDGES * C_F;
  gcn_scatter<<<(unsigned)((sc2 + T - 1) / T), T, 0, stream>>>(EI, dinv, hw, out, C_F, sc2);
  gcn_combine<<<(unsigned)((n2 + T - 1) / T), T, 0, stream>>>(dinv, hw, b2, out, C_F, 0, n2);
}